// GCN_5832565588575
// MI455X (gfx1250) — compile-verified
//
#include <hip/hip_runtime.h>

#define IN_CH 128
#define HID_CH 32
#define OUT_CH 16

typedef __attribute__((ext_vector_type(2))) float v2f;
typedef __attribute__((ext_vector_type(8))) float v8f;

// ---------------------------------------------------------------------------
// fp32 global atomic add: prefer hardware global_atomic_add_f32
// ---------------------------------------------------------------------------
__device__ __forceinline__ void atomic_fadd(float* p, float v) {
#if __has_builtin(__builtin_amdgcn_global_atomic_fadd_f32)
    __builtin_amdgcn_global_atomic_fadd_f32(
        (__attribute__((address_space(1))) float*)p, v);
#else
    atomicAdd(p, v);
#endif
}

// ---------------------------------------------------------------------------
// Utility: zero a float buffer
// ---------------------------------------------------------------------------
__global__ void k_zero(float* __restrict__ p, int n) {
    int i = blockIdx.x * blockDim.x + threadIdx.x;
    if (i < n) p[i] = 0.0f;
}

// ---------------------------------------------------------------------------
// In-degree count: deg[dst[e]] += 1
// ---------------------------------------------------------------------------
__global__ void k_deg(const long long* __restrict__ dst, float* __restrict__ deg, int E) {
    int e = blockIdx.x * blockDim.x + threadIdx.x;
    if (e < E) {
        atomic_fadd(&deg[(int)dst[e]], 1.0f);
    }
}

// ---------------------------------------------------------------------------
// dinv = deg > 0 ? deg^-0.5 : 0   (in place)
// ---------------------------------------------------------------------------
__global__ void k_dinv(float* __restrict__ deg, int n) {
    int i = blockIdx.x * blockDim.x + threadIdx.x;
    if (i < n) {
        float d = deg[i];
        deg[i] = (d > 0.0f) ? rsqrtf(d) : 0.0f;
    }
}

// ---------------------------------------------------------------------------
// Broadcast bias into accumulation buffer: out[n][c] = b[c]  (C = power of 2)
// ---------------------------------------------------------------------------
template <int C>
__global__ void k_init_bias(float* __restrict__ out, const float* __restrict__ b,
                            int total) {
    int i = blockIdx.x * blockDim.x + threadIdx.x;
    if (i < total) out[i] = b[i & (C - 1)];
}

// ---------------------------------------------------------------------------
// GEMM1: H[N,32] = X[N,128] @ W1[128,32]  via V_WMMA_F32_16X16X4_F32
// One wave -> one 16-row tile, two 16-col accumulators.
// W1 staged transposed in LDS (stride 130 -> 8B-aligned ds_load_b64 B-frags).
// ---------------------------------------------------------------------------
#define W1_STRIDE (IN_CH + 2)   // 130: keeps col*stride even, varies LDS bank
__global__ __launch_bounds__(256) void k_gemm1(const float* __restrict__ X,
                                               const float* __restrict__ W1,
                                               float* __restrict__ H,
                                               int N, int nTiles) {
    __shared__ __align__(16) float sWT[HID_CH * W1_STRIDE];
    int tid = threadIdx.x;
    for (int i = tid; i < IN_CH * HID_CH; i += 256) {
        int k = i >> 5;          // 0..127
        int col = i & 31;        // 0..31
        sWT[col * W1_STRIDE + k] = W1[i];
    }
    __syncthreads();

    int wave = tid >> 5;
    int lane = tid & 31;
    int tile = blockIdx.x * 8 + wave;
    if (tile >= nTiles) return;            // wave-uniform: EXEC stays full

    int mlane = lane & 15;
    int khalf = (lane >> 4) << 1;          // 0 for lanes 0-15, 2 for lanes 16-31
    int arow  = tile * 16 + mlane;
    if (arow >= N) arow = N - 1;           // clamp: garbage only hits unstored rows
    const float* xrow = X + (size_t)arow * IN_CH + khalf;
    const float* w0 = &sWT[mlane * W1_STRIDE + khalf];
    const float* w1 = &sWT[(16 + mlane) * W1_STRIDE + khalf];

    v8f c0 = {};
    v8f c1 = {};
#pragma unroll 8
    for (int k = 0; k < IN_CH; k += 4) {
        v2f a  = *(const v2f*)(xrow + k);  // aligned global b64
        v2f b0 = *(const v2f*)(w0 + k);    // aligned ds b64
        v2f b1 = *(const v2f*)(w1 + k);
        c0 = __builtin_amdgcn_wmma_f32_16x16x4_f32(false, a, false, b0,
                                                   (short)0, c0, false, false);
        c1 = __builtin_amdgcn_wmma_f32_16x16x4_f32(false, a, false, b1,
                                                   (short)0, c1, false, false);
    }

    // C/D layout: VGPR v, lane l -> row = v + 8*(l>=16), col = l&15
    int rbase = tile * 16 + ((lane >> 4) << 3);
    float* hp = H + (size_t)rbase * HID_CH + mlane;
    if (tile * 16 + 16 <= N) {
        // full tile: one base address, 16 stores with immediate offsets
#pragma unroll
        for (int v = 0; v < 8; ++v) {
            hp[v * HID_CH]      = c0[v];
            hp[v * HID_CH + 16] = c1[v];
        }
    } else {
#pragma unroll
        for (int v = 0; v < 8; ++v) {
            if (rbase + v < N) {
                hp[v * HID_CH]      = c0[v];
                hp[v * HID_CH + 16] = c1[v];
            }
        }
    }
}

// ---------------------------------------------------------------------------
// Aggregation layer 1: out1[dst] += H[src] * (dinv[src]*dinv[dst])
// One thread per edge: indices/norm loaded once, 8 float4 quads.
// ---------------------------------------------------------------------------
__global__ void k_agg1(const long long* __restrict__ src,
                       const long long* __restrict__ dst,
                       const float* __restrict__ dinv,
                       const float* __restrict__ H,
                       float* __restrict__ out1, int E) {
    int e = blockIdx.x * blockDim.x + threadIdx.x;
    if (e >= E) return;
    int s = (int)src[e];
    int d = (int)dst[e];
    float w = dinv[s] * dinv[d];
    const float4* hp = (const float4*)(H + (size_t)s * HID_CH);
    float* o = out1 + (size_t)d * HID_CH;
#pragma unroll
    for (int q = 0; q < HID_CH / 4; ++q) {
        float4 h = hp[q];
        atomic_fadd(o + 4 * q + 0, h.x * w);
        atomic_fadd(o + 4 * q + 1, h.y * w);
        atomic_fadd(o + 4 * q + 2, h.z * w);
        atomic_fadd(o + 4 * q + 3, h.w * w);
    }
}

// ---------------------------------------------------------------------------
// GEMM2: H2[N,16] = relu(out1[N,32]) @ W2[32,16]  (ReLU fused into A load)
// ---------------------------------------------------------------------------
#define W2_STRIDE (HID_CH + 2)  // 34: keeps col*stride even
__global__ __launch_bounds__(256) void k_gemm2(const float* __restrict__ A,
                                               const float* __restrict__ W2,
                                               float* __restrict__ H2,
                                               int N, int nTiles) {
    __shared__ __align__(16) float sWT[OUT_CH * W2_STRIDE];
    int tid = threadIdx.x;
    for (int i = tid; i < HID_CH * OUT_CH; i += 256) {
        int k = i >> 4;          // 0..31
        int col = i & 15;        // 0..15
        sWT[col * W2_STRIDE + k] = W2[i];
    }
    __syncthreads();

    int wave = tid >> 5;
    int lane = tid & 31;
    int tile = blockIdx.x * 8 + wave;
    if (tile >= nTiles) return;

    int mlane = lane & 15;
    int khalf = (lane >> 4) << 1;
    int arow  = tile * 16 + mlane;
    if (arow >= N) arow = N - 1;
    const float* ar = A + (size_t)arow * HID_CH + khalf;
    const float* wp = &sWT[mlane * W2_STRIDE + khalf];

    v8f c0 = {};
#pragma unroll
    for (int k = 0; k < HID_CH; k += 4) {
        v2f a = *(const v2f*)(ar + k);
        a.x = fmaxf(a.x, 0.0f);            // fused ReLU
        a.y = fmaxf(a.y, 0.0f);
        v2f b = *(const v2f*)(wp + k);
        c0 = __builtin_amdgcn_wmma_f32_16x16x4_f32(false, a, false, b,
                                                   (short)0, c0, false, false);
    }

    int rbase = tile * 16 + ((lane >> 4) << 3);
    float* hp = H2 + (size_t)rbase * OUT_CH + mlane;
    if (tile * 16 + 16 <= N) {
#pragma unroll
        for (int v = 0; v < 8; ++v) {
            hp[v * OUT_CH] = c0[v];
        }
    } else {
#pragma unroll
        for (int v = 0; v < 8; ++v) {
            if (rbase + v < N) {
                hp[v * OUT_CH] = c0[v];
            }
        }
    }
}

// ---------------------------------------------------------------------------
// Aggregation layer 2: out[dst] += H2[src] * norm   (one thread per edge)
// ---------------------------------------------------------------------------
__global__ void k_agg2(const long long* __restrict__ src,
                       const long long* __restrict__ dst,
                       const float* __restrict__ dinv,
                       const float* __restrict__ H2,
                       float* __restrict__ out, int E) {
    int e = blockIdx.x * blockDim.x + threadIdx.x;
    if (e >= E) return;
    int s = (int)src[e];
    int d = (int)dst[e];
    float w = dinv[s] * dinv[d];
    const float4* hp = (const float4*)(H2 + (size_t)s * OUT_CH);
    float* o = out + (size_t)d * OUT_CH;
#pragma unroll
    for (int q = 0; q < OUT_CH / 4; ++q) {
        float4 h = hp[q];
        atomic_fadd(o + 4 * q + 0, h.x * w);
        atomic_fadd(o + 4 * q + 1, h.y * w);
        atomic_fadd(o + 4 * q + 2, h.z * w);
        atomic_fadd(o + 4 * q + 3, h.w * w);
    }
}

// ---------------------------------------------------------------------------
// Host-side launch
// ---------------------------------------------------------------------------
extern "C" void kernel_launch(void* const* d_in, const int* in_sizes, int n_in,
                              void* d_out, int out_size, void* d_ws, size_t ws_size,
                              hipStream_t stream) {
    const float*     X  = (const float*)d_in[0];
    const long long* EI = (const long long*)d_in[1];
    const float*     W1 = (const float*)d_in[2];
    const float*     b1 = (const float*)d_in[3];
    const float*     W2 = (const float*)d_in[4];
    const float*     b2 = (const float*)d_in[5];
    float* out = (float*)d_out;

    const int N = in_sizes[0] / IN_CH;       // 100000
    const int E = in_sizes[1] / 2;           // 1600000
    const long long* src = EI;               // edge_index[0, :]
    const long long* dst = EI + E;           // edge_index[1, :]

    // Workspace layout (float elements), offsets 16B-aligned
    size_t nAl = ((size_t)N + 3) & ~(size_t)3;
    float* dinv = (float*)d_ws;              // N
    float* H    = dinv + nAl;                // N*32
    float* out1 = H    + nAl * HID_CH;       // N*32
    float* H2   = out1 + nAl * HID_CH;       // N*16

    const int nTiles = (N + 15) / 16;
    const int gemmBlocks = (nTiles + 7) / 8;

    // 1) degree -> dinv
    k_zero<<<(N + 255) / 256, 256, 0, stream>>>(dinv, N);
    k_deg<<<(E + 255) / 256, 256, 0, stream>>>(dst, dinv, E);
    k_dinv<<<(N + 255) / 256, 256, 0, stream>>>(dinv, N);

    // 2) layer 1: GEMM + (bias-init, scatter-add)
    k_gemm1<<<gemmBlocks, 256, 0, stream>>>(X, W1, H, N, nTiles);
    k_init_bias<HID_CH><<<((size_t)N * HID_CH + 255) / 256, 256, 0, stream>>>(out1, b1, N * HID_CH);
    k_agg1<<<(E + 255) / 256, 256, 0, stream>>>(src, dst, dinv, H, out1, E);

    // 3) layer 2: GEMM (fused ReLU on A) + (bias-init, scatter-add)
    k_gemm2<<<gemmBlocks, 256, 0, stream>>>(out1, W2, H2, N, nTiles);
    k_init_bias<OUT_CH><<<((size_t)N * OUT_CH + 255) / 256, 256, 0, stream>>>(out, b2, N * OUT_CH);
    k_agg2<<<(E + 255) / 256, 256, 0, stream>>>(src, dst, dinv, H2, out, E);
}